// NpuOp_npu_roi_align_Module_63015760167076
// MI455X (gfx1250) — compile-verified
//
#include <hip/hip_runtime.h>
#include <hip/hip_bf16.h>

// ROI Align for MI455X (gfx1250).
// feat: (B=2, C=256, H=200, W=200) f32, rois: (N=1000, 5) f32
// out:  (N, C, 7, 7) f32
//
// Memory-bound gather kernel. Design choices (see analysis):
//  - 1 thread per output element, pw-fastest layout -> coalesced writes,
//    row-local gathers within a wave.
//  - n is wave-uniform (49*256 = 12544 outputs per ROI, multiple of 32):
//    readfirstlane forces ROI params onto the scalar path (s_load).
//  - Non-temporal output stores keep the 82 MB feature map resident in the
//    192 MB L2 (output is write-once, never re-read).

#define RA_C   256
#define RA_H   200
#define RA_W   200
#define RA_PH  7
#define RA_PW  7
#define RA_SCALE 0.25f

__device__ __forceinline__ void bilin1d(float c, int size,
                                        int& lo, int& hi,
                                        float& wl, float& wh, bool& valid) {
    // Matches reference _bilinear_1d exactly.
    valid = (c >= -1.0f) && (c <= (float)size);
    float cc = fminf(fmaxf(c, 0.0f), (float)(size - 1));
    int l = (int)floorf(cc);
    l = min(max(l, 0), size - 1);
    lo = l;
    hi = min(l + 1, size - 1);
    float frac = cc - (float)l;
    wl = 1.0f - frac;
    wh = frac;
}

__global__ __launch_bounds__(256)
void roi_align_fwd_kernel(const float* __restrict__ feat,
                          const float* __restrict__ rois,
                          float* __restrict__ out,
                          int total) {
    int idx = blockIdx.x * blockDim.x + threadIdx.x;
    if (idx >= total) return;

    const int pw = idx % RA_PW;
    const int ph = (idx / RA_PW) % RA_PH;
    const int c  = (idx / (RA_PW * RA_PH)) % RA_C;
    int n        = idx / (RA_PW * RA_PH * RA_C);

    // n is uniform across the wave (12544 outputs per ROI, 12544 % 32 == 0).
    // Force it scalar so the 5 ROI loads become s_load (KMcnt path) instead
    // of per-lane vector loads.
    n = __builtin_amdgcn_readfirstlane(n);

    const float* __restrict__ r = rois + n * 5;
    const int   b  = (int)r[0];
    const float x1 = r[1] * RA_SCALE;
    const float y1 = r[2] * RA_SCALE;
    const float x2 = (r[3] + 1.0f) * RA_SCALE;   // ROI_END_MODE = 1
    const float y2 = (r[4] + 1.0f) * RA_SCALE;

    const float roi_w = fmaxf(x2 - x1, 1.0f);
    const float roi_h = fmaxf(y2 - y1, 1.0f);
    const float bw = roi_w / 7.0f;
    const float bh = roi_h / 7.0f;

    const float* __restrict__ fb = feat + (size_t)(b * RA_C + c) * (RA_H * RA_W);

    // Precompute the two x sample positions for this bin (shared by both iy).
    int   xl[2], xh[2];
    float wxl[2], wxh[2];
    bool  vx[2];
#pragma unroll
    for (int ix = 0; ix < 2; ++ix) {
        const float g  = (float)(2 * pw + ix) + 0.5f;
        const float xc = x1 + (bw * g) * 0.5f;      // start + bin*(g+0.5)/SN
        bilin1d(xc, RA_W, xl[ix], xh[ix], wxl[ix], wxh[ix], vx[ix]);
    }

    float acc = 0.0f;
#pragma unroll
    for (int iy = 0; iy < 2; ++iy) {
        const float g  = (float)(2 * ph + iy) + 0.5f;
        const float yc = y1 + (bh * g) * 0.5f;
        int yl, yh;
        float wyl, wyh;
        bool vy;
        bilin1d(yc, RA_H, yl, yh, wyl, wyh, vy);

        const float* __restrict__ row0 = fb + yl * RA_W;
        const float* __restrict__ row1 = fb + yh * RA_W;

#pragma unroll
        for (int ix = 0; ix < 2; ++ix) {
            if (vy && vx[ix]) {
                // 4 gathered corners; lanes in the wave walk along the row.
                const float v00 = row0[xl[ix]];
                const float v01 = row0[xh[ix]];
                const float v10 = row1[xl[ix]];
                const float v11 = row1[xh[ix]];
                acc += wyl * (wxl[ix] * v00 + wxh[ix] * v01)
                     + wyh * (wxl[ix] * v10 + wxh[ix] * v11);
            }
        }
    }

    // Mean over the 2x2 samples. Non-temporal store: output is write-once,
    // keep L2 for the feature map.
    __builtin_nontemporal_store(acc * 0.25f, &out[idx]);
}

extern "C" void kernel_launch(void* const* d_in, const int* in_sizes, int n_in,
                              void* d_out, int out_size, void* d_ws, size_t ws_size,
                              hipStream_t stream) {
    const float* feat = (const float*)d_in[0];
    const float* rois = (const float*)d_in[1];
    float* out = (float*)d_out;

    const int total = out_size;           // N * 256 * 7 * 7
    const int threads = 256;              // 8 wave32 waves per block
    const int blocks = (total + threads - 1) / threads;

    roi_align_fwd_kernel<<<blocks, threads, 0, stream>>>(feat, rois, out, total);
}